// CPD_30245159698617
// MI455X (gfx1250) — compile-verified
//
#include <hip/hip_runtime.h>

// CP inner-product sampling: out[i] = sum_r f0[i0,r]*f1[i1,r]*f2[i2,r], RANK=32.
// Memory-bound gather kernel; the three factor matrices (38 MB total) are
// L2-resident on MI455X (192 MB L2), so steady-state traffic is indices + out.
// WMMA f32 16x16x4 with an all-ones B operand is the rank-reduction engine:
// D[m][n] += sum_k P[m][k] for all n, i.e. every column of D carries the
// row-sums, computed on the matrix pipe in full f32 (RNE) like the reference.

typedef __attribute__((ext_vector_type(2))) float v2f;
typedef __attribute__((ext_vector_type(4))) float v4f;
typedef __attribute__((ext_vector_type(8))) float v8f;

#define RANK 32
#define WAVES_PER_BLOCK 8   // 256 threads = 8 wave32

__global__ __launch_bounds__(256) void cpd_sampled_dot_wmma(
    const int* __restrict__ idxs,   // [nnz][3] int32
    const float* __restrict__ f0,   // [N0][32]
    const float* __restrict__ f1,   // [N1][32]
    const float* __restrict__ f2,   // [N2][32]
    float* __restrict__ out,        // [nnz]
    int nnz)
{
    const int lane = threadIdx.x & 31;
    const int waveInBlock = threadIdx.x >> 5;
    const int tile = blockIdx.x * WAVES_PER_BLOCK + waveInBlock; // 16 nnz / tile
    const int base = tile * 16;
    if (base >= nnz) return;              // wave-uniform exit (EXEC stays all-1s)

    const int m  = lane & 15;             // tile row (nnz) owned by this lane
    const int hi = lane >> 4;             // 0: K pair {0,1}; 1: K pair {2,3}

    int gi = base + m;
    if (gi >= nnz) gi = nnz - 1;          // clamp (nnz % 16 == 0 in practice)

    const int i0 = idxs[gi * 3 + 0];
    const int i1 = idxs[gi * 3 + 1];
    const int i2 = idxs[gi * 3 + 2];

    const float* __restrict__ r0 = f0 + (size_t)i0 * RANK;
    const float* __restrict__ r1 = f1 + (size_t)i1 * RANK;
    const float* __restrict__ r2 = f2 + (size_t)i2 * RANK;

    // All-ones B (4x16): every column of D becomes the row-sum of A.
    v2f ones; ones[0] = 1.0f; ones[1] = 1.0f;
    v8f acc = {};

    const int k0 = hi * 2;                // lane's K offset inside the 4-chunk
    #pragma unroll
    for (int kk = 0; kk < RANK; kk += 4) {
        const int k = kk + k0;
        // two consecutive rank elements per factor -> b64 loads, L2-resident
        const float a0 = r0[k]     * r1[k]     * r2[k];
        const float a1 = r0[k + 1] * r1[k + 1] * r2[k + 1];
        v2f a; a[0] = a0; a[1] = a1;
        // A(16x4 f32) x B(4x16 ones) accumulated into C(16x16 f32)
        acc = __builtin_amdgcn_wmma_f32_16x16x4_f32(
            /*neg_a=*/false, a, /*neg_b=*/false, ones,
            /*c_mod=*/(short)0, acc, /*reuse_a=*/false, /*reuse_b=*/false);
    }

    // C/D f32 16x16 layout: lanes 0-15 hold M = vgpr (0..7) at N = lane;
    // lanes 16-31 hold M = vgpr + 8. All N columns are identical (ones-B),
    // so lane 0 owns rowsums 0..7 and lane 16 owns rowsums 8..15.
    if (m == 0) {
        float* dst = out + base + hi * 8;          // 32B-aligned (base % 16 == 0)
        if (base + 16 <= nnz) {
            // Full tile (always true for nnz % 16 == 0): two b128 stores.
            v4f lo, hiq;
            lo[0] = acc[0]; lo[1] = acc[1]; lo[2] = acc[2]; lo[3] = acc[3];
            hiq[0] = acc[4]; hiq[1] = acc[5]; hiq[2] = acc[6]; hiq[3] = acc[7];
            *(v4f*)(dst)     = lo;
            *(v4f*)(dst + 4) = hiq;
        } else {
            #pragma unroll
            for (int j = 0; j < 8; ++j) {
                if (base + hi * 8 + j < nnz) dst[j] = acc[j];
            }
        }
    }
}

extern "C" void kernel_launch(void* const* d_in, const int* in_sizes, int n_in,
                              void* d_out, int out_size, void* d_ws, size_t ws_size,
                              hipStream_t stream)
{
    const int*   idxs = (const int*)  d_in[0];   // [nnz][3] int32
    const float* f0   = (const float*)d_in[1];
    const float* f1   = (const float*)d_in[2];
    const float* f2   = (const float*)d_in[3];
    float*       out  = (float*)d_out;

    const int nnz    = in_sizes[0] / 3;
    const int tiles  = (nnz + 15) / 16;
    const int blocks = (tiles + WAVES_PER_BLOCK - 1) / WAVES_PER_BLOCK;

    cpd_sampled_dot_wmma<<<blocks, 256, 0, stream>>>(idxs, f0, f1, f2, out, nnz);
}